// GraphLearner_IB_89137751261400
// MI455X (gfx1250) — compile-verified
//
#include <hip/hip_runtime.h>

#define B_DIM 2048
#define N_DIM 100
#define H_DIM 512
#define F_DIM 1536

typedef __attribute__((ext_vector_type(2))) float v2f;
typedef __attribute__((ext_vector_type(4))) float v4f;
typedef __attribute__((ext_vector_type(8))) float v8f;

// ---------------------------------------------------------------------------
// Kernel 1: streaming mean-pool over N=100 for both base tensors.
// Pure bandwidth (~839 MB read). Non-temporal loads keep the one-shot stream
// from thrashing L2 (weights/activations stay resident there).
// ---------------------------------------------------------------------------
__global__ __launch_bounds__(256) void mean_pool_kernel(
    const float* __restrict__ base_img, const float* __restrict__ base_txt,
    float* __restrict__ mean_img, float* __restrict__ mean_txt) {
  const int HV = H_DIM / 4;  // 128 float4 per row
  int gid = blockIdx.x * blockDim.x + threadIdx.x;  // one thread per (b, h4)
  int b = gid / HV;
  int hv = gid - b * HV;
  const v4f* ip = (const v4f*)(base_img + (size_t)b * N_DIM * H_DIM) + hv;
  const v4f* tp = (const v4f*)(base_txt + (size_t)b * N_DIM * H_DIM) + hv;
  v4f si = (v4f)0.0f;
  v4f st = (v4f)0.0f;
#pragma unroll 4
  for (int n = 0; n < N_DIM; ++n) {
    si += __builtin_nontemporal_load(ip + (size_t)n * HV);
    st += __builtin_nontemporal_load(tp + (size_t)n * HV);
  }
  const float inv = 1.0f / (float)N_DIM;
  *((v4f*)(mean_img + (size_t)b * H_DIM) + hv) = si * inv;
  *((v4f*)(mean_txt + (size_t)b * H_DIM) + hv) = st * inv;
}

// ---------------------------------------------------------------------------
// fp32 WMMA helper: accumulate a 16(M) x 64(N) tile over K using
// V_WMMA_F32_16X16X4_F32.  out[m][n] += sum_k A[m][k] * W[n][koff + k]
// (i.e. A @ W^T, which is exactly the reference's x @ W.T convention).
//
// Per ISA layouts (wave32):
//   A 16x4 : lane l<16 -> {A[l][k0],A[l][k0+1]}, lane 16+l -> {k0+2,k0+3}
//   B 4x16 : lane l<16 -> {W[n0+l][k0],W[n0+l][k0+1]}, lane 16+l -> {k0+2,k0+3}
//   C/D    : v8f, element r = row (hi*8 + r), col n0 + l
// ---------------------------------------------------------------------------
__device__ __forceinline__ void gemm_acc_16x64(
    v8f acc[4],
    const float* __restrict__ A, int lda,
    const float* __restrict__ W, int ldw, int koff, int K,
    int m0, int n0, int l, int hi) {
  const float* arow = A + (size_t)(m0 + l) * lda + 2 * hi;
  const float* wrow = W + (size_t)(n0 + l) * ldw + koff + 2 * hi;
#pragma unroll 2
  for (int k = 0; k < K; k += 4) {
    v2f a = *(const v2f*)(arow + k);
#pragma unroll
    for (int j = 0; j < 4; ++j) {
      v2f b = *(const v2f*)(wrow + (size_t)(j * 16) * ldw + k);
      acc[j] = __builtin_amdgcn_wmma_f32_16x16x4_f32(
          /*neg_a=*/false, a, /*neg_b=*/false, b,
          /*c_mod=*/(short)0, acc[j], /*reuse_a=*/false, /*reuse_b=*/false);
    }
  }
}

// ---------------------------------------------------------------------------
// Kernel 2: u = concat(text, img, compress) @ W_user^T + b_user
// K = 1536 split into three 512-wide segments, one per source tensor.
// Wave tile 16x64, block = 4 waves -> 64x64, grid (32, 8).
// ---------------------------------------------------------------------------
__global__ __launch_bounds__(128) void user_linear_kernel(
    const float* __restrict__ in_text, const float* __restrict__ in_img,
    const float* __restrict__ in_comp, const float* __restrict__ W_user,
    const float* __restrict__ b_user, float* __restrict__ u) {
  const int lane = threadIdx.x & 31;
  const int wave = threadIdx.x >> 5;
  const int l = lane & 15;
  const int hi = lane >> 4;
  const int m0 = blockIdx.x * 64 + wave * 16;
  const int n0 = blockIdx.y * 64;

  v8f acc[4] = {};
  gemm_acc_16x64(acc, in_text, 512, W_user, F_DIM,    0, 512, m0, n0, l, hi);
  gemm_acc_16x64(acc, in_img,  512, W_user, F_DIM,  512, 512, m0, n0, l, hi);
  gemm_acc_16x64(acc, in_comp, 512, W_user, F_DIM, 1024, 512, m0, n0, l, hi);

#pragma unroll
  for (int j = 0; j < 4; ++j) {
    const int col = n0 + j * 16 + l;
    const float bias = b_user[col];
#pragma unroll
    for (int r = 0; r < 8; ++r) {
      u[(size_t)(m0 + hi * 8 + r) * H_DIM + col] = acc[j][r] + bias;
    }
  }
}

// ---------------------------------------------------------------------------
// Kernel 3: out = relu(mean_img @ Wl_img^T + u @ Wr_img^T
//                    + mean_txt @ Wl_txt^T + u @ Wr_txt^T + bl_img + bl_txt)
// Four K=512 GEMM passes accumulate into the same WMMA accumulators.
// ---------------------------------------------------------------------------
__global__ __launch_bounds__(128) void sage_out_kernel(
    const float* __restrict__ mean_img, const float* __restrict__ mean_txt,
    const float* __restrict__ u,
    const float* __restrict__ Wl_img, const float* __restrict__ Wr_img,
    const float* __restrict__ Wl_txt, const float* __restrict__ Wr_txt,
    const float* __restrict__ bl_img, const float* __restrict__ bl_txt,
    float* __restrict__ out) {
  const int lane = threadIdx.x & 31;
  const int wave = threadIdx.x >> 5;
  const int l = lane & 15;
  const int hi = lane >> 4;
  const int m0 = blockIdx.x * 64 + wave * 16;
  const int n0 = blockIdx.y * 64;

  v8f acc[4] = {};
  gemm_acc_16x64(acc, mean_img, H_DIM, Wl_img, H_DIM, 0, H_DIM, m0, n0, l, hi);
  gemm_acc_16x64(acc, u,        H_DIM, Wr_img, H_DIM, 0, H_DIM, m0, n0, l, hi);
  gemm_acc_16x64(acc, mean_txt, H_DIM, Wl_txt, H_DIM, 0, H_DIM, m0, n0, l, hi);
  gemm_acc_16x64(acc, u,        H_DIM, Wr_txt, H_DIM, 0, H_DIM, m0, n0, l, hi);

#pragma unroll
  for (int j = 0; j < 4; ++j) {
    const int col = n0 + j * 16 + l;
    const float bias = bl_img[col] + bl_txt[col];
#pragma unroll
    for (int r = 0; r < 8; ++r) {
      float v = acc[j][r] + bias;
      out[(size_t)(m0 + hi * 8 + r) * H_DIM + col] = fmaxf(v, 0.0f);
    }
  }
}

// ---------------------------------------------------------------------------
extern "C" void kernel_launch(void* const* d_in, const int* in_sizes, int n_in,
                              void* d_out, int out_size, void* d_ws, size_t ws_size,
                              hipStream_t stream) {
  const float* input_text     = (const float*)d_in[0];   // [B,1,512]
  const float* input_img      = (const float*)d_in[1];   // [B,1,512]
  const float* input_compress = (const float*)d_in[2];   // [B,512]
  const float* base_txt       = (const float*)d_in[3];   // [B,N,H]
  const float* base_img       = (const float*)d_in[4];   // [B,N,H]
  const float* W_user         = (const float*)d_in[5];   // [H,F]
  const float* b_user         = (const float*)d_in[6];   // [H]
  const float* Wl_img         = (const float*)d_in[7];   // [H,H]
  const float* bl_img         = (const float*)d_in[8];   // [H]
  const float* Wr_img         = (const float*)d_in[9];   // [H,H]
  const float* Wl_txt         = (const float*)d_in[10];  // [H,H]
  const float* bl_txt         = (const float*)d_in[11];  // [H]
  const float* Wr_txt         = (const float*)d_in[12];  // [H,H]
  float* out = (float*)d_out;

  float* ws = (float*)d_ws;
  const size_t BH = (size_t)B_DIM * H_DIM;
  float* mean_img = ws;           // [B,H]
  float* mean_txt = ws + BH;      // [B,H]
  float* u        = ws + 2 * BH;  // [B,H]

  // 1) bandwidth-bound mean pools (independent of GEMM below)
  mean_pool_kernel<<<(B_DIM * (H_DIM / 4)) / 256, 256, 0, stream>>>(
      base_img, base_txt, mean_img, mean_txt);

  // 2) user linear (2048x512x1536 fp32 WMMA GEMM)
  user_linear_kernel<<<dim3(B_DIM / 64, H_DIM / 64), 128, 0, stream>>>(
      input_text, input_img, input_compress, W_user, b_user, u);

  // 3) fused 4-way SAGE GEMM + bias + ReLU
  sage_out_kernel<<<dim3(B_DIM / 64, H_DIM / 64), 128, 0, stream>>>(
      mean_img, mean_txt, u, Wl_img, Wr_img, Wl_txt, Wr_txt,
      bl_img, bl_txt, out);
}